// CoDAConv2d_19911468384354
// MI455X (gfx1250) — compile-verified
//
#include <hip/hip_runtime.h>

typedef __attribute__((ext_vector_type(2))) float v2f;
typedef __attribute__((ext_vector_type(8))) float v8f;
typedef __attribute__((ext_vector_type(4))) unsigned int u32x4;
typedef __attribute__((ext_vector_type(8))) unsigned int u32x8;

namespace {
constexpr int KS   = 3;
constexpr int CIN  = 8;
constexpr int COUT = 16;
constexpr int PATCH = CIN * KS * KS;         // 72
constexpr int HH = 112, WW = 112, BB = 4;
constexpr int SEG_W = 16;                    // pixels per wave (WMMA N dim)
constexpr int SEGS_PER_ROW = WW / SEG_W;     // 7
constexpr int NSEG = BB * HH * SEGS_PER_ROW; // 3136
constexpr int WAVES = 8;                     // waves per block (256 threads)
constexpr int LDSW = PATCH * COUT * CIN;     // 9216 floats of w_pred
constexpr int LDSB = PATCH * COUT;           // 1152 floats of b_pred
}

// Issue one TDM descriptor: copy n_elems consecutive f32 from global -> LDS.
// D# layout per CDNA5 ISA ch.8 (async_tensor): group0 = {count, lds_addr,
// global_addr[56:0], type=2}; group1 = {mask=0, data_size=2 (4B), 1-row 2D
// tile: tensor_dim0 = tile_dim0 = n_elems, tensor_dim1 = tile_dim1 = 1}.
__device__ __forceinline__ void tdm_load_to_lds(const void* gsrc,
                                                unsigned lds_byte_off,
                                                unsigned n_elems) {
  unsigned long long ga = (unsigned long long)gsrc;
  u32x4 g0;
  g0[0] = 1u;                                         // count=1, user D#, no gather
  g0[1] = lds_byte_off;                               // lds_addr (bytes)
  g0[2] = (unsigned)(ga & 0xFFFFFFFFu);               // global_addr[31:0]
  g0[3] = (unsigned)((ga >> 32) & 0x01FFFFFFu)        // global_addr[56:32]
          | (2u << 30);                               // type = 2 ("image")
  u32x8 g1;
  g1[0] = (2u << 16);                                 // data_size=4B, wg_mask=0
  g1[1] = (n_elems & 0xFFFFu) << 16;                  // tensor_dim0[15:0] @63:48
  g1[2] = ((n_elems >> 16) & 0xFFFFu) | (1u << 16);   // tensor_dim0[31:16], tensor_dim1=1
  g1[3] = (n_elems & 0xFFFFu) << 16;                  // tile_dim0 @127:112
  g1[4] = 1u;                                         // tile_dim1=1, tile_dim2=0
  g1[5] = n_elems;                                    // tensor_dim0_stride[31:0]
  g1[6] = 0u;                                         // stride0 hi, stride1 lo
  g1[7] = 0u;                                         // stride1 hi
  // 2-group form: VADDR2/VADDR3 = NULL (tensor rank <= 2).
  asm volatile("tensor_load_to_lds %0, %1" :: "s"(g0), "s"(g1) : "memory");
}

__global__ __launch_bounds__(256, 1)
void coda_conv2d_wmma(const float* __restrict__ in,
                      const float* __restrict__ wpred,
                      const float* __restrict__ bpred,
                      float* __restrict__ out) {
  // Stage dynamic-weight predictor matrix + bias in LDS via the TDM (one
  // issuing wave; others release at the barrier once data has landed).
  __shared__ __align__(32) float lds[LDSW + LDSB];
  const int tid = threadIdx.x;
  if (tid < 32) {
    unsigned base = (unsigned)(unsigned long long)(uintptr_t)&lds[0];
    tdm_load_to_lds(wpred, base, (unsigned)LDSW);
    tdm_load_to_lds(bpred, base + LDSW * 4u, (unsigned)LDSB);
    __builtin_amdgcn_s_wait_tensorcnt(0);
  }
  __syncthreads();

  const int lane = tid & 31;
  const int n    = lane & 15;   // WMMA column N (= pixel in segment) and A-row M
  const int half = lane >> 4;   // lane half: K pair for A/B, +8 row offset for C/D
  const int seg  = blockIdx.x * WAVES + (tid >> 5);   // < NSEG (392*8 == 3136)
  const int b    = seg / (HH * SEGS_PER_ROW);
  const int rem  = seg - b * (HH * SEGS_PER_ROW);
  const int h    = rem / SEGS_PER_ROW;
  const int x    = (rem - h * SEGS_PER_ROW) * SEG_W + n;

  const size_t plane = (size_t)HH * WW;

  // ---- B matrix (4x16 f32, K x N): X[c][n] = in[b, c, h, x] ----
  const float* px = in + (size_t)b * CIN * plane + (size_t)h * WW + x;
  const int ck = half * 2;
  v2f bm0, bm1;
  bm0.x = px[(size_t)(ck + 0) * plane];
  bm0.y = px[(size_t)(ck + 1) * plane];
  bm1.x = px[(size_t)(ck + 4) * plane];
  bm1.y = px[(size_t)(ck + 5) * plane];

  v8f nrm = {};   // sum_p w^2 per (co, pixel)
  v8f act = {};   // sum_p patch*w per (co, pixel)

  const float* ldsw = lds;
  const float* ldsb = lds + LDSW;

  int p = 0;
#pragma unroll 1
  for (int c = 0; c < CIN; ++c) {
    const float* cbase = in + ((size_t)b * CIN + c) * plane;
#pragma unroll
    for (int ky = 0; ky < KS; ++ky) {
      const int hy  = h + ky - 1;
      const bool hok = (hy >= 0) && (hy < HH);
      const int hyc = hy < 0 ? 0 : (hy >= HH ? HH - 1 : hy);
      const float* rbase = cbase + (size_t)hyc * WW;
#pragma unroll
      for (int kx = 0; kx < KS; ++kx, ++p) {
        // A tile (16x4 f32): rows o = p*16 + M, M = n; K pair selected by half.
        const float* arow = ldsw + (size_t)(p * 16 + n) * CIN + ck;
        v2f a0 = *(const v2f*)arow;         // K = ck, ck+1   (channels 0..3 step)
        v2f a1 = *(const v2f*)(arow + 4);   // K = ck+4, ck+5 (channels 4..7 step)

        // C accumulator pre-loaded with bias (uniform across N):
        // lane holds M = vgpr + 8*half -> 8 consecutive bias values.
        v8f cmat = *(const v8f*)(ldsb + p * 16 + half * 8);

        // raw[p*16+M][n] = W @ X + bias, via two K=4 WMMA steps.
        v8f d = __builtin_amdgcn_wmma_f32_16x16x4_f32(
            false, a0, false, bm0, (short)0, cmat, false, false);
        d = __builtin_amdgcn_wmma_f32_16x16x4_f32(
            false, a1, false, bm1, (short)0, d, false, false);

        // Patch value per-column (per-lane): zero-padded, but loaded
        // unconditionally at a clamped (always in-bounds) address + select,
        // so no EXEC-mask branching in the hot loop.
        const int wx  = x + kx - 1;
        const int wxc = wx < 0 ? 0 : (wx >= WW ? WW - 1 : wx);
        float pv = rbase[wxc];
        pv = (hok && wx >= 0 && wx < WW) ? pv : 0.0f;

        nrm += d * d;     // norm accumulation over patch dim
        act += pv * d;    // weighted patch sum
      }
    }
  }

  // out[b, co, h, x] = act / (sqrt(nrm) + 1e-6); co = half*8 + vgpr.
  float* obase = out + ((size_t)b * COUT + half * 8) * plane + (size_t)h * WW + x;
#pragma unroll
  for (int v = 0; v < 8; ++v) {
    float denom = __builtin_sqrtf(nrm[v]) + 1e-6f;
    obase[(size_t)v * plane] = act[v] / denom;
  }
}

extern "C" void kernel_launch(void* const* d_in, const int* in_sizes, int n_in,
                              void* d_out, int out_size, void* d_ws, size_t ws_size,
                              hipStream_t stream) {
  (void)in_sizes; (void)n_in; (void)out_size; (void)d_ws; (void)ws_size;
  const float* in    = (const float*)d_in[0];
  const float* wpred = (const float*)d_in[1];
  const float* bpred = (const float*)d_in[2];
  float* out = (float*)d_out;

  dim3 grid(NSEG / WAVES);   // 392 blocks
  dim3 block(256);           // 8 waves of 32
  coda_conv2d_wmma<<<grid, block, 0, stream>>>(in, wpred, bpred, out);
}